// BayesianBatchNorm_11261404250602
// MI455X (gfx1250) — compile-verified
//
#include <hip/hip_runtime.h>
#include <stdint.h>

// BayesianBatchNorm for MI455X (gfx1250, wave32).
// Memory-bound: ~617 MB total traffic -> ~26us at 23.3 TB/s. Kernel 1 uses the
// Tensor Data Mover (TDM) to DMA channel tiles into LDS (double buffered,
// s_wait_tensorcnt), and a v_wmma_f32_16x16x4_f32 with an all-ones B matrix as
// a layout-safe 32-lane reduction. Kernel 3 streams with non-temporal stores.

#define N_  64
#define C_  256
#define H_  56
#define W_  56
#define HW_ (H_ * W_)                 // 3136
#define NELEM ((size_t)N_ * C_ * HW_) // 51380224
#define NPERC ((float)(N_ * HW_))     // 200704

#define ROWS_PER_STAGE 2
#define NSTAGES (N_ / ROWS_PER_STAGE)          // 32
#define STAGE_FLOATS (ROWS_PER_STAGE * HW_)    // 6272
#define MOM 0.1f
#define EPSN 1e-5f
#define JIT 1e-5f

typedef __attribute__((ext_vector_type(2))) float v2f;
typedef __attribute__((ext_vector_type(4))) float v4f;
typedef __attribute__((ext_vector_type(8))) float v8f;
typedef __attribute__((ext_vector_type(4))) unsigned int v4u;
typedef __attribute__((ext_vector_type(4))) int v4i;
typedef __attribute__((ext_vector_type(8))) int v8i;

// Issue one TDM load: 2D tile (ROWS_PER_STAGE rows x HW_ floats), row stride
// C_*HW_ elements, into LDS at byte offset lds_off. D# per CDNA5 ISA §8.3-8.6.
// This toolchain exposes the 6-arg builtin:
//   (v4u group0, v8i group1, v4i group2, v4i group3, v8i extra, i32 cpol)
// groups 2/3 and the trailing group are zero for a 2D tensor.
__device__ __forceinline__ void tdm_issue(const float* gptr, unsigned lds_off) {
  unsigned long long ga = (unsigned long long)(uintptr_t)gptr;
  v4u g0;
  g0[0] = 1u;                                  // count=1, user descriptor
  g0[1] = lds_off;                             // lds_addr (bytes)
  g0[2] = (unsigned)(ga & 0xFFFFFFFFu);        // global_addr[31:0]
  g0[3] = (unsigned)((ga >> 32) & 0x01FFFFFFu) // global_addr[56:32]
          | (2u << 30);                        // type = 2 ("image")
  const unsigned long long stride0 = (unsigned long long)C_ * HW_; // elements
  v8i g1;
  g1[0] = (int)(2u << 16);                              // data_size = 4 bytes
  g1[1] = (int)((unsigned)(HW_ & 0xFFFF) << 16);        // tensor_dim0[15:0]
  g1[2] = (int)(((unsigned)HW_ >> 16) | ((unsigned)N_ << 16)); // dim0 hi | tensor_dim1 lo
  g1[3] = (int)(((unsigned)N_ >> 16) | ((unsigned)(HW_ & 0xFFFF) << 16)); // dim1 hi | tile_dim0
  g1[4] = (int)(unsigned)ROWS_PER_STAGE;                // tile_dim1 | tile_dim2=0
  g1[5] = (int)(unsigned)(stride0 & 0xFFFFFFFFull);     // tensor_dim0_stride lo
  g1[6] = (int)(unsigned)((stride0 >> 32) & 0xFFFFull); // stride hi | dim1_stride=0
  g1[7] = 0;
  v4i gz = {0, 0, 0, 0};                                // groups 2/3: 2D tensor
  v8i gz8 = {0, 0, 0, 0, 0, 0, 0, 0};
  __builtin_amdgcn_tensor_load_to_lds(g0, g1, gz, gz, gz8, 0);
}

// Kernel 1: one block per channel; TDM-staged double-buffered reduction.
__global__ void __launch_bounds__(256) bbn_stats_kernel(
    const float* __restrict__ in, float* __restrict__ ws_sum,
    float* __restrict__ ws_sq) {
  __shared__ __align__(16) float buf[2][STAGE_FLOATS];
  __shared__ float red[8][2];
  const int c = blockIdx.x;
  const int tid = threadIdx.x;
  const float* chan = in + (size_t)c * HW_; // row n lives at + n*C_*HW_
  const unsigned lds0 = (unsigned)(uintptr_t)&buf[0][0];
  const unsigned lds1 = (unsigned)(uintptr_t)&buf[1][0];

  if (tid < 32) tdm_issue(chan, lds0); // prime stage 0 -> buf0

  float s = 0.f, q = 0.f;
  for (int st = 0; st < NSTAGES; ++st) {
    if (tid < 32) {
      if (st + 1 < NSTAGES) {
        const float* nxt = chan + (size_t)(st + 1) * ROWS_PER_STAGE * C_ * HW_;
        tdm_issue(nxt, (st & 1) ? lds0 : lds1); // next stage -> other buffer
        __builtin_amdgcn_s_wait_tensorcnt(1);   // in-order: stage st complete
      } else {
        __builtin_amdgcn_s_wait_tensorcnt(0);
      }
    }
    __syncthreads();
    const v4f* b4 = (const v4f*)&buf[st & 1][0];
    for (int i = tid; i < STAGE_FLOATS / 4; i += 256) {
      v4f v = b4[i];
      s += (v[0] + v[1]) + (v[2] + v[3]);
      q = fmaf(v[0], v[0], fmaf(v[1], v[1], fmaf(v[2], v[2], fmaf(v[3], v[3], q))));
    }
    __syncthreads();
  }

  // Cross-lane reduce via WMMA: D = A x ones, A is the documented 16x4 f32
  // striping (lane L, vgpr j -> A[L%16, 2*(L/16)+j]).  With A=(partial,0),
  // row m of D = partial[m] + partial[m+16], replicated over all 16 columns.
  // Each lane then sums its 8 C/D vgprs = rows {0..7} (lanes 0-15) or rows
  // {8..15} (lanes 16-31); one shfl_xor(16) finishes the 32-lane sum.
  v2f a0; a0[0] = s; a0[1] = 0.f;
  v2f a1; a1[0] = q; a1[1] = 0.f;
  v2f ones; ones[0] = 1.f; ones[1] = 1.f;
  v8f z = {};
  v8f dsum = __builtin_amdgcn_wmma_f32_16x16x4_f32(false, a0, false, ones,
                                                   (short)0, z, false, false);
  v8f dsq = __builtin_amdgcn_wmma_f32_16x16x4_f32(false, a1, false, ones,
                                                  (short)0, z, false, false);
  float ps = ((dsum[0] + dsum[1]) + (dsum[2] + dsum[3])) +
             ((dsum[4] + dsum[5]) + (dsum[6] + dsum[7]));
  float pq = ((dsq[0] + dsq[1]) + (dsq[2] + dsq[3])) +
             ((dsq[4] + dsq[5]) + (dsq[6] + dsq[7]));
  ps += __shfl_xor(ps, 16, 32);
  pq += __shfl_xor(pq, 16, 32);
  const int wv = tid >> 5;
  if ((tid & 31) == 0) { red[wv][0] = ps; red[wv][1] = pq; }
  __syncthreads();
  if (tid == 0) {
    float S = 0.f, Q = 0.f;
    for (int w = 0; w < 8; ++w) { S += red[w][0]; Q += red[w][1]; }
    ws_sum[c] = S;
    ws_sq[c] = Q;
  }
}

// Kernel 2: per-channel running stats + scale/shift + symmetric-KL scalar.
__global__ void __launch_bounds__(256) bbn_finalize_kernel(
    const float* __restrict__ ws_sum, const float* __restrict__ ws_sq,
    const float* __restrict__ layer_rm, const float* __restrict__ layer_rv,
    const float* __restrict__ lw, const float* __restrict__ lb,
    const float* __restrict__ norm_rm, const float* __restrict__ norm_rv,
    const float* __restrict__ prior, float* __restrict__ ws_scale,
    float* __restrict__ ws_shift, float* __restrict__ div_out) {
  __shared__ float sdiv[256];
  const int c = threadIdx.x;
  const float n = NPERC;
  float mean = ws_sum[c] / n;
  float var = ws_sq[c] / n - mean * mean;
  var = fmaxf(var, 0.f);
  float var_u = var * (n / (n - 1.f));
  float rm_t = (1.f - MOM) * norm_rm[c] + MOM * mean;
  float rv_t = (1.f - MOM) * norm_rv[c] + MOM * var_u;

  float mu_s = layer_rm[c];
  float v_s = layer_rv[c] + JIT;
  float v_t = rv_t + JIT;
  float d = rm_t - mu_s;
  // symmetric KL: log terms cancel; per-channel contribution:
  sdiv[c] = 0.25f * (v_s / v_t + v_t / v_s + d * d * (1.f / v_t + 1.f / v_s) - 2.f);

  float p = prior[0];
  float running_mean = p * mu_s + (1.f - p) * rm_t;
  float running_var =
      p * layer_rv[c] + (1.f - p) * rv_t + p * (1.f - p) * d * d;
  float sc = lw[c] / sqrtf(running_var + EPSN);
  ws_scale[c] = sc;
  ws_shift[c] = lb[c] - running_mean * sc;

  __syncthreads();
  for (int off = 128; off > 0; off >>= 1) {
    if (c < off) sdiv[c] += sdiv[c + off];
    __syncthreads();
  }
  if (c == 0) *div_out = sdiv[0];
}

// Kernel 3: streaming affine apply, float4, non-temporal stores.
__global__ void __launch_bounds__(256) bbn_apply_kernel(
    const float* __restrict__ in, const float* __restrict__ ws_scale,
    const float* __restrict__ ws_shift, float* __restrict__ out) {
  __shared__ float sc[C_], sh[C_];
  sc[threadIdx.x] = ws_scale[threadIdx.x];
  sh[threadIdx.x] = ws_shift[threadIdx.x];
  __syncthreads();
  const long long total4 = (long long)(NELEM / 4); // 12,845,056
  const v4f* in4 = (const v4f*)in;
  v4f* out4 = (v4f*)out;
  for (long long i = (long long)blockIdx.x * blockDim.x + threadIdx.x;
       i < total4; i += (long long)gridDim.x * blockDim.x) {
    const int cidx = (int)((i / (HW_ / 4)) & (C_ - 1)); // chunk = n*C + c
    v4f v = in4[i];
    const float a = sc[cidx], b = sh[cidx];
    v4f r;
    r[0] = fmaf(v[0], a, b);
    r[1] = fmaf(v[1], a, b);
    r[2] = fmaf(v[2], a, b);
    r[3] = fmaf(v[3], a, b);
    __builtin_nontemporal_store(r, &out4[i]); // output never re-read
  }
}

extern "C" void kernel_launch(void* const* d_in, const int* in_sizes, int n_in,
                              void* d_out, int out_size, void* d_ws,
                              size_t ws_size, hipStream_t stream) {
  (void)in_sizes; (void)n_in; (void)out_size; (void)ws_size;
  const float* input    = (const float*)d_in[0];
  const float* layer_rm = (const float*)d_in[1];
  const float* layer_rv = (const float*)d_in[2];
  const float* lw       = (const float*)d_in[3];
  const float* lb       = (const float*)d_in[4];
  const float* norm_rm  = (const float*)d_in[5];
  const float* norm_rv  = (const float*)d_in[6];
  const float* prior    = (const float*)d_in[7];
  float* out = (float*)d_out;
  float* ws = (float*)d_ws;
  float* ws_sum = ws;            // [C]
  float* ws_sq = ws + C_;        // [C]
  float* ws_scale = ws + 2 * C_; // [C]
  float* ws_shift = ws + 3 * C_; // [C]
  float* div_out = out + NELEM;  // tuple: (out, div) concatenated

  bbn_stats_kernel<<<C_, 256, 0, stream>>>(input, ws_sum, ws_sq);
  bbn_finalize_kernel<<<1, 256, 0, stream>>>(ws_sum, ws_sq, layer_rm, layer_rv,
                                             lw, lb, norm_rm, norm_rv, prior,
                                             ws_scale, ws_shift, div_out);
  bbn_apply_kernel<<<4096, 256, 0, stream>>>(input, ws_scale, ws_shift, out);
}